// ProximityAwareLoss4Class_25786983645938
// MI455X (gfx1250) — compile-verified
//
#include <hip/hip_runtime.h>
#include <hip/hip_bf16.h>

// ProximityAwareLoss4Class for MI455X (gfx1250, wave32).
// One 1024-thread workgroup per batch row; bitmask-window distance transform
// (MAX_DISTANCE=10) entirely in LDS; V_WMMA_F32_16X16X4_F32 used for the
// f32-exact per-wave partial-sum reduction.

typedef float v2f __attribute__((ext_vector_type(2)));
typedef float v8f __attribute__((ext_vector_type(8)));

#define S_LEN   16384
#define NWORDS  (S_LEN / 32)     // 512 mask words per row
#define BLOCK   1024
#define NWAVES  (BLOCK / 32)     // 32 wave32s
#define ITERS   (S_LEN / BLOCK)  // 16 positions per thread

// Distance (0..10) from bit `lane` of word `widx` to the nearest set bit in
// `mask` within +-10 positions; returns 1000 if none in the window. Matches
// the reference's min-distance transform wherever the result is used
// (exact d for d<=10, predicate for d>10).
__device__ __forceinline__ int window_dist(const unsigned* mask, int widx, int lane) {
    unsigned w0 = mask[widx];
    unsigned wm = (widx > 0)          ? mask[widx - 1] : 0u;
    unsigned wp = (widx < NWORDS - 1) ? mask[widx + 1] : 0u;
    unsigned long long lo = ((unsigned long long)w0 << 32) | (unsigned long long)wm;
    unsigned long long hi = ((unsigned long long)wp << 32) | (unsigned long long)w0;
    int c = 32 + lane;                                    // combined bit index of p
    unsigned left11  = (unsigned)(lo >> (c - 10)) & 0x7FFu;   // bits p-10..p
    unsigned right10 = (unsigned)(hi >> (lane + 1)) & 0x3FFu; // bits p+1..p+10
    if (left11 & 0x400u) return 0;                        // bit at p itself
    int d = 1000;
    unsigned lb = left11 & 0x3FFu;                        // bit j -> distance 10-j
    if (lb) d = 10 - (31 - __builtin_clz(lb));
    if (right10) {                                        // bit j -> distance j+1
        int dr = __builtin_ctz(right10) + 1;
        if (dr < d) d = dr;
    }
    return d;
}

__global__ __launch_bounds__(BLOCK)
void ProximityAwareLoss4Class_25786983645938_kernel(const float* __restrict__ logits,
                                                    const int*   __restrict__ labels,
                                                    float*       __restrict__ ws) {
    __shared__ float    s_ce[S_LEN];        // 64 KiB: per-position weighted CE
    __shared__ unsigned s_sw[NWORDS];       // true-switch bitmask
    __shared__ unsigned s_psw[NWORDS];      // predicted-switch bitmask
    __shared__ unsigned s_val[NWORDS];      // valid bitmask
    __shared__ float    s_dw[11];           // distance weights
    __shared__ int      s_anyp;             // has_pred (row-wide any)
    __shared__ float    s_rs[NWAVES], s_rc[NWAVES];

    const int b    = blockIdx.x;
    const int tid  = threadIdx.x;
    const int lane = tid & 31;
    const int wave = tid >> 5;

    if (tid < 11) {
        const float dwi[11] = {1.0f, 0.99f, 0.95f, 0.9f, 0.85f, 0.8f,
                               0.75f, 0.7f, 0.65f, 0.6f, 0.55f};
        s_dw[tid] = dwi[tid];
    }
    if (tid == 0) s_anyp = 0;
    __syncthreads();

    const float4* __restrict__ lg = (const float4*)logits + (size_t)b * S_LEN;
    const int*    __restrict__ lb = labels + (size_t)b * S_LEN;

    float lsum = 0.0f;   // accumulates fp-penalty now, (ce2 - bonus) in phase 2
    float lcnt = 0.0f;   // valid count

    // -------- Phase 1: stream logits/labels once (coalesced b128 loads) -----
    for (int k = 0; k < ITERS; ++k) {
        const int widx = k * NWAVES + wave;   // wave owns whole 32-bit mask word
        const int p    = widx * 32 + lane;    // lanes cover 32 consecutive pos
        if (k + 1 < ITERS) __builtin_prefetch(&lg[p + BLOCK], 0, 3); // global_prefetch_b8

        float4 x  = lg[p];
        int    lab = lb[p];
        bool valid = (lab != -100);
        int  l     = valid ? lab : 0;

        float m = x.x; int am = 0;            // argmax = first max (jnp semantics)
        if (x.y > m) { m = x.y; am = 1; }
        if (x.z > m) { m = x.z; am = 2; }
        if (x.w > m) { m = x.w; am = 3; }

        float se  = __expf(x.x - m) + __expf(x.y - m) + __expf(x.z - m) + __expf(x.w - m);
        float lse = m + __logf(se);
        float xl  = (l == 0) ? x.x : (l == 1) ? x.y : (l == 2) ? x.z : x.w;
        float cw  = (l >= 2) ? 10.0f : 1.0f;  // class_weights = {1,1,10,10}
        float ce  = valid ? (lse - xl) * cw : 0.0f;

        bool sw  = (lab == 2) || (lab == 3);
        bool psw = (am  == 2) || (am  == 3);

        s_ce[p] = ce;
        unsigned msw  = __builtin_amdgcn_ballot_w32(sw);
        unsigned mpsw = __builtin_amdgcn_ballot_w32(psw);
        unsigned mval = __builtin_amdgcn_ballot_w32(valid);
        if (lane == 0) {
            s_sw[widx] = msw; s_psw[widx] = mpsw; s_val[widx] = mval;
            if (mpsw) s_anyp = 1;
        }
        if (valid) {
            lcnt += 1.0f;
            if (psw && !sw) lsum += 2.0f;     // FALSE_POS_PENALTY
        }
    }
    __syncthreads();

    const bool anyp = (s_anyp != 0);

    // -------- Phase 2: LDS-only bonus/penalty via +-10 window queries -------
    for (int k = 0; k < ITERS; ++k) {
        const int widx = k * NWAVES + wave;
        const int p    = widx * 32 + lane;
        bool valid = (s_val[widx] >> lane) & 1u;
        if (!valid) continue;
        bool sw  = (s_sw[widx]  >> lane) & 1u;
        bool psw = (s_psw[widx] >> lane) & 1u;
        float ce = s_ce[p];
        float ce2 = ce, bonus = 0.0f;
        if (psw) {                                       // d<=10 implies has_true
            int d = window_dist(s_sw, widx, lane);
            if (d <= 10) bonus = ce * s_dw[d] * 0.5f;    // BONUS_SCALE
        }
        if (sw && anyp) {
            int d = window_dist(s_psw, widx, lane);
            if (d > 10) ce2 = ce * 2.0f;                 // MISS_MULT
        }
        lsum += ce2 - bonus;
    }

    // -------- Block reduction: V_WMMA_F32_16X16X4_F32 with all-ones B -------
    // D[m][n] = sum_k A[m][k]; with B==1 and summing ALL 256 D entries the
    // result is 16 * (sum of all A entries) regardless of lane layout.
    // EXEC is all-ones here (uniform control flow, full block).
    v2f a;    a[0] = lsum; a[1] = 0.0f;
    v2f ones; ones[0] = 1.0f; ones[1] = 1.0f;
    v8f cz = {0.f, 0.f, 0.f, 0.f, 0.f, 0.f, 0.f, 0.f};
    v8f d = __builtin_amdgcn_wmma_f32_16x16x4_f32(false, a, false, ones,
                                                  (short)0, cz, false, false);
    float s = d[0] + d[1] + d[2] + d[3] + d[4] + d[5] + d[6] + d[7];
    for (int off = 16; off; off >>= 1) s += __shfl_xor(s, off, 32);
    float waveSum = s * 0.0625f;   // /16: each row-sum replicated over 16 cols

    float c = lcnt;
    for (int off = 16; off; off >>= 1) c += __shfl_xor(c, off, 32);

    if (lane == 0) { s_rs[wave] = waveSum; s_rc[wave] = c; }
    __syncthreads();
    if (tid == 0) {
        float Sb = 0.0f, Cb = 0.0f;
        for (int i = 0; i < NWAVES; ++i) { Sb += s_rs[i]; Cb += s_rc[i]; }
        ws[2 * b]     = Sb;
        ws[2 * b + 1] = Cb;
    }
}

__global__ void ProximityAwareLoss4Class_finalize(const float* __restrict__ ws,
                                                  float* __restrict__ out,
                                                  int nblocks) {
    __shared__ float ss[8], sc[8];
    int t = threadIdx.x;  // 256 threads, 8 waves
    float s = 0.0f, c = 0.0f;
    if (t < nblocks) { s = ws[2 * t]; c = ws[2 * t + 1]; }
    for (int off = 16; off; off >>= 1) {
        s += __shfl_xor(s, off, 32);
        c += __shfl_xor(c, off, 32);
    }
    if ((t & 31) == 0) { ss[t >> 5] = s; sc[t >> 5] = c; }
    __syncthreads();
    if (t == 0) {
        float S = 0.0f, C = 0.0f;
        for (int i = 0; i < 8; ++i) { S += ss[i]; C += sc[i]; }
        out[0] = S / fmaxf(C, 1.0f);
    }
}

extern "C" void kernel_launch(void* const* d_in, const int* in_sizes, int n_in,
                              void* d_out, int out_size, void* d_ws, size_t ws_size,
                              hipStream_t stream) {
    const float* logits = (const float*)d_in[0];
    const int*   labels = (const int*)d_in[1];   // harness passes integer inputs as int32
    float* out = (float*)d_out;
    float* ws  = (float*)d_ws;

    const int B = in_sizes[1] / S_LEN;           // labels flat count = B*S
    ProximityAwareLoss4Class_25786983645938_kernel<<<B, BLOCK, 0, stream>>>(logits, labels, ws);
    ProximityAwareLoss4Class_finalize<<<1, 256, 0, stream>>>(ws, out, B);
}